// FTRL_FM_28999619182790
// MI455X (gfx1250) — compile-verified
//
#include <hip/hip_runtime.h>

// FTRL-FM constants (match reference)
#define B_SAMPLES 4096
#define K_FEATS   50
#define M_DIM     32
#define WAVES_PER_BLOCK 4
#define BLOCK_THREADS   (WAVES_PER_BLOCK * 32)
#define ROWS_PAD  52   // 13 groups * 4 rows (50 real rows + 2 pad rows)

// Pointer types matching the gfx1250 async-to-LDS builtin signature:
//   void __builtin_amdgcn_global_load_async_to_lds_b128(v4i as(1)*, v4i as(3)*, Ii, Ii)
typedef int v4i __attribute__((vector_size(16)));
typedef __attribute__((address_space(1))) v4i gv4i;   // global int4
typedef __attribute__((address_space(3))) v4i lv4i;   // LDS int4

__device__ __forceinline__ void async_ld_b128(const void* gsrc, void* ldst) {
  // per-lane 16B global -> LDS DMA, tracked by ASYNCcnt
  __builtin_amdgcn_global_load_async_to_lds_b128(
      (gv4i*)gsrc, (lv4i*)ldst, /*imm offset*/0, /*cpol*/0);
}

// FTRL lazy weight: 0 inside L1 ball, else (sign(z)*ss - z)/((beta+sqrt(n))/alpha + l2)
// alpha=0.1, beta=1.0 -> denom = 10*sqrt(n) + (10 + l2).
// n ∈ [0.1, 10.1] so raw v_sqrt_f32 / v_rcp_f32 (1 TRANS op each) are safe.
__device__ __forceinline__ float ftrl_w(float z, float n, float l1, float l2, float ss) {
  float sgn   = (z >= 0.0f) ? ss : -ss;
  float denom = __builtin_fmaf(__builtin_amdgcn_sqrtf(n), 10.0f, 10.0f + l2);
  float w     = (sgn - z) * __builtin_amdgcn_rcpf(denom);
  return (__builtin_fabsf(z) <= l1) ? 0.0f : w;
}

__global__ __launch_bounds__(BLOCK_THREADS)
void ftrl_fm_kernel(const int* __restrict__ x_idx,
                    const float* __restrict__ z1t,
                    const float* __restrict__ n1t,
                    const float* __restrict__ z2t,
                    const float* __restrict__ n2t,
                    float* __restrict__ out)
{
  __shared__ int   sh_idx[WAVES_PER_BLOCK][64];
  __shared__ float sh_z[WAVES_PER_BLOCK][ROWS_PAD * M_DIM];
  __shared__ float sh_n[WAVES_PER_BLOCK][ROWS_PAD * M_DIM];

  const int lane = threadIdx.x & 31;
  const int wv   = threadIdx.x >> 5;
  const int b    = blockIdx.x * WAVES_PER_BLOCK + wv;   // one wave per sample

  // ---- stage feature indices (nothing else on the DMA critical path) ----
  const int base = b * K_FEATS;
  int idxA = x_idx[base + lane];                 // k = lane (0..31)
  sh_idx[wv][lane] = idxA;
  int idxB = idxA;                               // safe default for lanes >= 18
  if (lane < K_FEATS - 32) {                     // k = 32 + lane (32..49)
    idxB = x_idx[base + 32 + lane];
    sh_idx[wv][32 + lane] = idxB;
  }

  __syncthreads();   // sh_idx visible before address generation

  // ---- issue all async row gathers FIRST: 26 B128 DMA ops per wave ----
  // lane layout per group: r = row-in-group (4 rows), sub = 16B chunk within 128B row
  const int r   = lane >> 3;
  const int sub = (lane & 7) * 16;
  #pragma unroll
  for (int g = 0; g < 13; ++g) {
    int k  = 4 * g + r;
    int kk = (k < K_FEATS) ? k : (K_FEATS - 1);  // pad rows duplicate a valid row
    long f = (long)sh_idx[wv][kk];
    const char* gz = (const char*)z2t + f * 128 + sub;
    const char* gn = (const char*)n2t + f * 128 + sub;
    char* lz = (char*)&sh_z[wv][0] + (long)k * 128 + sub;
    char* ln = (char*)&sh_n[wv][0] + (long)k * 128 + sub;
    async_ld_b128(gz, lz);
    async_ld_b128(gn, ln);
  }

  // ---- first-order term: scalar gathers overlap the in-flight DMA ----
  float linp = ftrl_w(z1t[idxA], n1t[idxA], 0.05f, 0.05f, 0.05f);
  if (lane < K_FEATS - 32)
    linp += ftrl_w(z1t[idxB], n1t[idxB], 0.05f, 0.05f, 0.05f);

  // ---- two-stage pipeline: async loads complete in order ----
  float s = 0.0f, ssq = 0.0f;

  // Stage 1: first 12 DMA ops (groups 0..5 -> rows 0..23) have landed
  __builtin_amdgcn_s_wait_asynccnt(14);
  __asm__ volatile("" ::: "memory");
  #pragma unroll
  for (int k = 0; k < 24; ++k) {
    float z = sh_z[wv][k * M_DIM + lane];
    float n = sh_n[wv][k * M_DIM + lane];
    float w = ftrl_w(z, n, 0.05f, 1.0f, 0.05f);
    s   += w;
    ssq += w * w;
  }

  // Stage 2: remaining rows
  __builtin_amdgcn_s_wait_asynccnt(0);
  __asm__ volatile("" ::: "memory");
  #pragma unroll
  for (int k = 24; k < K_FEATS; ++k) {
    float z = sh_z[wv][k * M_DIM + lane];
    float n = sh_n[wv][k * M_DIM + lane];
    float w = ftrl_w(z, n, 0.05f, 1.0f, 0.05f);
    s   += w;
    ssq += w * w;
  }

  // lin + 0.5*(||s||^2 - sum w^2): all terms are lane-linear -> single butterfly
  float u = linp + 0.5f * (s * s - ssq);
  #pragma unroll
  for (int off = 16; off >= 1; off >>= 1)
    u += __shfl_xor(u, off, 32);

  if (lane == 0) out[b] = u;
}

extern "C" void kernel_launch(void* const* d_in, const int* in_sizes, int n_in,
                              void* d_out, int out_size, void* d_ws, size_t ws_size,
                              hipStream_t stream) {
  (void)in_sizes; (void)n_in; (void)out_size; (void)d_ws; (void)ws_size;
  const int*   x_idx = (const int*)d_in[0];
  const float* z1    = (const float*)d_in[1];
  const float* n1    = (const float*)d_in[2];
  const float* z2    = (const float*)d_in[3];
  const float* n2    = (const float*)d_in[4];
  float* out = (float*)d_out;

  dim3 grid(B_SAMPLES / WAVES_PER_BLOCK);
  ftrl_fm_kernel<<<grid, BLOCK_THREADS, 0, stream>>>(x_idx, z1, n1, z2, n2, out);
}